// RNN_1683627180049
// MI455X (gfx1250) — compile-verified
//
#include <hip/hip_runtime.h>

// 2-layer tanh RNN, H=32, for MI455X (gfx1250, wave32).
// One wave handles 16 batch rows; all GEMMs via V_WMMA_F32_16X16X4_F32.

#define H 32

typedef __attribute__((ext_vector_type(2))) float v2f;
typedef __attribute__((ext_vector_type(8))) float v8f;

__device__ __forceinline__ float fast_tanh(float xx) {
  // tanh(x) = 1 - 2/(exp(2x)+1); exp2/rcp are HW transcendentals on CDNA5.
  float e = __builtin_amdgcn_exp2f(xx * 2.8853900817779268f); // 2*log2(e)
  return 1.0f - 2.0f * __builtin_amdgcn_rcpf(e + 1.0f);
}

__launch_bounds__(32)
__global__ void rnn2_wmma_kernel(const float* __restrict__ x,       // [B,T,1]
                                 const float* __restrict__ h_state, // [2,B,H]
                                 const float* __restrict__ W_ih0,   // [H,1]
                                 const float* __restrict__ W_hh0,   // [H,H]
                                 const float* __restrict__ b_ih0,   // [H]
                                 const float* __restrict__ b_hh0,   // [H]
                                 const float* __restrict__ W_ih1,   // [H,H]
                                 const float* __restrict__ W_hh1,   // [H,H]
                                 const float* __restrict__ b_ih1,   // [H]
                                 const float* __restrict__ b_hh1,   // [H]
                                 const float* __restrict__ W_out,   // [1,H]
                                 const float* __restrict__ b_out,   // [1]
                                 float* __restrict__ out,           // [B*T] ++ [2,B,H]
                                 int B, int T)
{
  // 16x32 state matrices, row stride 33 floats (bank-conflict padding).
  __shared__ float lds0[16 * 33];
  __shared__ float lds1[16 * 33];

  const int lane = threadIdx.x & 31;
  const int nl   = lane & 15;   // N (or row) sub-index within a 16-wide tile
  const int hi   = lane >> 4;   // lane half: selects K pair / M half

  const int b_base = blockIdx.x * 16;

  // ---- Weights in WMMA B-operand layout -------------------------------
  // GEMM: D[m,n] += sum_k A[m,k] * Wt[k,n], Wt = W^T, so B chunk element
  // (klocal, nlocal) of chunk c, tile j = W[(16j+nl)*H + (4c + 2*hi + v)].
  v2f b0[2][8], b1i[2][8], b1h[2][8];
#pragma unroll
  for (int j = 0; j < 2; ++j) {
    const int n = 16 * j + nl;
#pragma unroll
    for (int c = 0; c < 8; ++c) {
      const int k = 4 * c + 2 * hi;
      v2f t0; t0.x = W_hh0[n * H + k]; t0.y = W_hh0[n * H + k + 1]; b0[j][c]  = t0;
      v2f t1; t1.x = W_ih1[n * H + k]; t1.y = W_ih1[n * H + k + 1]; b1i[j][c] = t1;
      v2f t2; t2.x = W_hh1[n * H + k]; t2.y = W_hh1[n * H + k + 1]; b1h[j][c] = t2;
    }
  }

  float wi0[2], bs0[2], bs1[2];
#pragma unroll
  for (int j = 0; j < 2; ++j) {
    const int n = 16 * j + nl;
    wi0[j] = W_ih0[n];
    bs0[j] = b_ih0[n] + b_hh0[n];
    bs1[j] = b_ih1[n] + b_hh1[n];
  }

  // Output projection weights gathered in A-layout order for this lane.
  float woutA[16];
#pragma unroll
  for (int c = 0; c < 8; ++c) {
    woutA[2 * c]     = W_out[4 * c + 2 * hi];
    woutA[2 * c + 1] = W_out[4 * c + 2 * hi + 1];
  }
  const float bout = b_out[0];

  // ---- Initial hidden state -> LDS -> A-layout registers ---------------
#pragma unroll
  for (int m = 0; m < 16; ++m) {
    lds0[m * 33 + lane] = h_state[(b_base + m) * H + lane];
    lds1[m * 33 + lane] = h_state[B * H + (b_base + m) * H + lane];
  }
  v2f a0[8], a1[8];
#pragma unroll
  for (int c = 0; c < 8; ++c) {
    const int k = 4 * c + 2 * hi;
    v2f t0; t0.x = lds0[nl * 33 + k]; t0.y = lds0[nl * 33 + k + 1]; a0[c] = t0;
    v2f t1; t1.x = lds1[nl * 33 + k]; t1.y = lds1[nl * 33 + k + 1]; a1[c] = t1;
  }

  // x rows this lane needs for its C-layout accumulator rows: m = v + 8*hi.
  const long xrow = (long)(b_base + 8 * hi) * (long)T;
  float xv[8];
#pragma unroll
  for (int v = 0; v < 8; ++v) xv[v] = x[xrow + (long)v * T];

#pragma unroll 1
  for (int t = 0; t < T; ++t) {
    float xc[8];
#pragma unroll
    for (int v = 0; v < 8; ++v) xc[v] = xv[v];
    if (t + 1 < T) {
#pragma unroll
      for (int v = 0; v < 8; ++v) xv[v] = x[xrow + (long)v * T + (t + 1)];
    }

    // -- Layer 1 recurrent term first: independent of this step's layer 0.
    v8f c1[2];
#pragma unroll
    for (int j = 0; j < 2; ++j) {
#pragma unroll
      for (int v = 0; v < 8; ++v) c1[j][v] = bs1[j];
#pragma unroll
      for (int c = 0; c < 8; ++c)
        c1[j] = __builtin_amdgcn_wmma_f32_16x16x4_f32(
            false, a1[c], false, b1h[j][c], (short)0, c1[j], false, false);
    }

    // -- Layer 0: C init = x*W_ih0 + (b_ih0+b_hh0), then += H0 @ W_hh0^T.
    v8f c0[2];
#pragma unroll
    for (int j = 0; j < 2; ++j) {
#pragma unroll
      for (int v = 0; v < 8; ++v) c0[j][v] = xc[v] * wi0[j] + bs0[j];
#pragma unroll
      for (int c = 0; c < 8; ++c)
        c0[j] = __builtin_amdgcn_wmma_f32_16x16x4_f32(
            false, a0[c], false, b0[j][c], (short)0, c0[j], false, false);
    }

    // h0' = tanh(c0); C-layout -> LDS (row-major, stride 33).
#pragma unroll
    for (int j = 0; j < 2; ++j)
#pragma unroll
      for (int v = 0; v < 8; ++v)
        lds0[(v + 8 * hi) * 33 + 16 * j + nl] = fast_tanh(c0[j][v]);

    // Re-read h0' in A layout (same wave: DS in-order, no barrier needed).
#pragma unroll
    for (int c = 0; c < 8; ++c) {
      const int k = 4 * c + 2 * hi;
      v2f ta; ta.x = lds0[nl * 33 + k]; ta.y = lds0[nl * 33 + k + 1]; a0[c] = ta;
    }

    // -- Layer 1 input term: += h0' @ W_ih1^T.
#pragma unroll
    for (int j = 0; j < 2; ++j)
#pragma unroll
      for (int c = 0; c < 8; ++c)
        c1[j] = __builtin_amdgcn_wmma_f32_16x16x4_f32(
            false, a0[c], false, b1i[j][c], (short)0, c1[j], false, false);

    // h1' = tanh(c1) -> LDS -> A layout; fuse output dot with the reload.
#pragma unroll
    for (int j = 0; j < 2; ++j)
#pragma unroll
      for (int v = 0; v < 8; ++v)
        lds1[(v + 8 * hi) * 33 + 16 * j + nl] = fast_tanh(c1[j][v]);

    float p = 0.0f;
#pragma unroll
    for (int c = 0; c < 8; ++c) {
      const int k = 4 * c + 2 * hi;
      v2f ta; ta.x = lds1[nl * 33 + k]; ta.y = lds1[nl * 33 + k + 1];
      a1[c] = ta;
      p += ta.x * woutA[2 * c] + ta.y * woutA[2 * c + 1];
    }
    // Each row's dot is split across lane pair (l, l+16).
    p += __shfl_xor(p, 16, 32);
    if (hi == 0)
      out[(long)(b_base + nl) * T + t] = p + bout;
  }

  // ---- h_final: [2,B,H] appended after the B*T outputs -----------------
  const long BT = (long)B * (long)T;
#pragma unroll
  for (int m = 0; m < 16; ++m) {
    out[BT + (long)(b_base + m) * H + lane]               = lds0[m * 33 + lane];
    out[BT + (long)B * H + (long)(b_base + m) * H + lane] = lds1[m * 33 + lane];
  }
}

extern "C" void kernel_launch(void* const* d_in, const int* in_sizes, int n_in,
                              void* d_out, int out_size, void* d_ws, size_t ws_size,
                              hipStream_t stream) {
  const float* x      = (const float*)d_in[0];
  const float* hst    = (const float*)d_in[1];
  const float* W_ih0  = (const float*)d_in[2];
  const float* W_hh0  = (const float*)d_in[3];
  const float* b_ih0  = (const float*)d_in[4];
  const float* b_hh0  = (const float*)d_in[5];
  const float* W_ih1  = (const float*)d_in[6];
  const float* W_hh1  = (const float*)d_in[7];
  const float* b_ih1  = (const float*)d_in[8];
  const float* b_hh1  = (const float*)d_in[9];
  const float* W_out  = (const float*)d_in[10];
  const float* b_outp = (const float*)d_in[11];

  const int B = in_sizes[1] / (2 * H);      // h_state = [2,B,H]
  const int T = in_sizes[0] / B;            // x = [B,T,1]
  const int blocks = B / 16;                // 16 batch rows per wave

  rnn2_wmma_kernel<<<blocks, 32, 0, stream>>>(
      x, hst, W_ih0, W_hh0, b_ih0, b_hh0,
      W_ih1, W_hh1, b_ih1, b_hh1, W_out, b_outp,
      (float*)d_out, B, T);
}